// GCN_2layer_22582938042901
// MI455X (gfx1250) — compile-verified
//
#include <hip/hip_runtime.h>
#include <hip/hip_bf16.h>

typedef __attribute__((ext_vector_type(2))) float v2f;
typedef __attribute__((ext_vector_type(4))) float v4f;
typedef __attribute__((ext_vector_type(8))) float v8f;

#define N_NODES 100000
#define N_EDGES 3200000
#define IN_F    128
#define HID     16
#define OUT_F   64

// ---------------------------------------------------------------------------
// init: deg=1 (self loop), zero layer-1 accumulator and d_out (layer-2 accum)
// ---------------------------------------------------------------------------
__global__ void gcn_init(float* __restrict__ deg, float* __restrict__ agg1,
                         float* __restrict__ out) {
    long long idx = (long long)blockIdx.x * blockDim.x + threadIdx.x;
    if (idx < N_NODES)                     deg[idx]  = 1.0f;   // self-loop
    if (idx < (long long)N_NODES * HID)    agg1[idx] = 0.0f;
    if (idx < (long long)N_NODES * OUT_F)  out[idx]  = 0.0f;
}

// ---------------------------------------------------------------------------
// degree accumulation over real edges (dst side)
// ---------------------------------------------------------------------------
__global__ void gcn_degree(const int* __restrict__ ei, float* __restrict__ deg) {
    int e = blockIdx.x * blockDim.x + threadIdx.x;
    if (e < N_EDGES) {
        int d = ei[N_EDGES + e];           // edge_index[1][e]
        atomicAdd(&deg[d], 1.0f);
    }
}

__global__ void gcn_rsqrt(float* __restrict__ deg) {
    int i = blockIdx.x * blockDim.x + threadIdx.x;
    if (i < N_NODES) {
        float d = deg[i];
        deg[i] = (d > 0.0f) ? rsqrtf(d) : 0.0f;   // becomes dinv
    }
}

// ---------------------------------------------------------------------------
// GEMM1: h1[N,16] = x[N,128] @ W1[128,16]  — V_WMMA_F32_16X16X4_F32
// One wave per 16-row tile, K stepped by 4 (32 WMMAs per tile).
// A 16x4 layout: lane l -> M=l&15, VGPR j holds K=(l>>4)*2+j
// B 4x16 layout: lane l -> N=l&15, VGPR j holds K=(l>>4)*2+j
// C/D layout:    VGPR r -> M=r+(l>>4)*8, N=l&15
// ---------------------------------------------------------------------------
__global__ void gcn_gemm1_wmma(const float* __restrict__ x,
                               const float* __restrict__ W1,
                               float* __restrict__ h1) {
    const int wave = blockIdx.x * (blockDim.x >> 5) + (threadIdx.x >> 5);
    const int lane = threadIdx.x & 31;
    const int ntiles = N_NODES / 16;               // 6250, exact
    if (wave >= ntiles) return;                    // wave-uniform: EXEC all-1s

    const int m0    = wave * 16;
    const int mlow  = lane & 15;
    const int khalf = (lane >> 4) * 2;
    const int mrow  = m0 + mlow;

    v8f acc = {};
    #pragma unroll
    for (int k0 = 0; k0 < IN_F; k0 += 4) {
        v2f a, b;
        a.x = x[(long long)mrow * IN_F + k0 + khalf];
        a.y = x[(long long)mrow * IN_F + k0 + khalf + 1];
        b.x = W1[(k0 + khalf)     * HID + mlow];
        b.y = W1[(k0 + khalf + 1) * HID + mlow];
        acc = __builtin_amdgcn_wmma_f32_16x16x4_f32(
                  false, a, false, b, (short)0, acc, false, false);
    }
    #pragma unroll
    for (int r = 0; r < 8; ++r) {
        int m = m0 + r + ((lane >> 4) << 3);
        h1[(long long)m * HID + mlow] = acc[r];
    }
}

// ---------------------------------------------------------------------------
// Edge scatter, layer 1: one thread per (edge, 4 features) -> b128 gather,
// 4 f32 atomics. E real edges + N self loops. h1/agg1 (6.4 MB) are
// L2-resident; atomics resolve in L2.
// ---------------------------------------------------------------------------
__global__ void gcn_scatter16(const int* __restrict__ ei,
                              const float* __restrict__ dinv,
                              const float* __restrict__ h,
                              float* __restrict__ agg) {
    long long idx = (long long)blockIdx.x * blockDim.x + threadIdx.x;
    const long long total = (long long)(N_EDGES + N_NODES) * (HID / 4);
    if (idx >= total) return;
    int f4 = (int)(idx & (HID / 4 - 1)) * 4;       // 0,4,8,12
    long long e = idx >> 2;
    int s, d;
    if (e < N_EDGES) { s = ei[e]; d = ei[N_EDGES + e]; }
    else             { s = d = (int)(e - N_EDGES); }
    float nrm = dinv[s] * dinv[d];
    v4f v = *(const v4f*)&h[(long long)s * HID + f4];
    float* dst = &agg[(long long)d * HID + f4];
    atomicAdd(dst + 0, v.x * nrm);
    atomicAdd(dst + 1, v.y * nrm);
    atomicAdd(dst + 2, v.z * nrm);
    atomicAdd(dst + 3, v.w * nrm);
}

// ---------------------------------------------------------------------------
// bias + ReLU, in place, vectorized x4 (F is 16 or 64, both /4)
// ---------------------------------------------------------------------------
__global__ void gcn_bias_relu(float* __restrict__ buf, const float* __restrict__ b,
                              int fmask, long long total4) {
    long long idx = (long long)blockIdx.x * blockDim.x + threadIdx.x;
    if (idx >= total4) return;
    int f = (int)((idx * 4) & fmask);
    v4f v = *(v4f*)&buf[idx * 4];
    v.x += b[f + 0]; v.y += b[f + 1]; v.z += b[f + 2]; v.w += b[f + 3];
    v.x = v.x > 0.0f ? v.x : 0.0f;
    v.y = v.y > 0.0f ? v.y : 0.0f;
    v.z = v.z > 0.0f ? v.z : 0.0f;
    v.w = v.w > 0.0f ? v.w : 0.0f;
    *(v4f*)&buf[idx * 4] = v;
}

// ---------------------------------------------------------------------------
// GEMM2: h3[N,64] = h1'[N,16] @ W2[16,64] — 16x16 tiles, 4 col-tiles, K=16
// ---------------------------------------------------------------------------
__global__ void gcn_gemm2_wmma(const float* __restrict__ h2,
                               const float* __restrict__ W2,
                               float* __restrict__ h3) {
    const int wave = blockIdx.x * (blockDim.x >> 5) + (threadIdx.x >> 5);
    const int lane = threadIdx.x & 31;
    const int rowTiles = N_NODES / 16;             // 6250
    const int colTiles = OUT_F / 16;               // 4
    if (wave >= rowTiles * colTiles) return;       // wave-uniform

    const int rt = wave >> 2;                      // / colTiles
    const int ct = wave & 3;                       // % colTiles
    const int m0 = rt * 16, c0 = ct * 16;
    const int mlow  = lane & 15;
    const int khalf = (lane >> 4) * 2;
    const int mrow  = m0 + mlow;

    v8f acc = {};
    #pragma unroll
    for (int k0 = 0; k0 < HID; k0 += 4) {
        v2f a, b;
        a.x = h2[(long long)mrow * HID + k0 + khalf];
        a.y = h2[(long long)mrow * HID + k0 + khalf + 1];
        b.x = W2[(k0 + khalf)     * OUT_F + c0 + mlow];
        b.y = W2[(k0 + khalf + 1) * OUT_F + c0 + mlow];
        acc = __builtin_amdgcn_wmma_f32_16x16x4_f32(
                  false, a, false, b, (short)0, acc, false, false);
    }
    #pragma unroll
    for (int r = 0; r < 8; ++r) {
        int m = m0 + r + ((lane >> 4) << 3);
        h3[(long long)m * OUT_F + c0 + mlow] = acc[r];
    }
}

// ---------------------------------------------------------------------------
// Edge scatter, layer 2: one thread per (edge, 4 features), into d_out
// ---------------------------------------------------------------------------
__global__ void gcn_scatter64(const int* __restrict__ ei,
                              const float* __restrict__ dinv,
                              const float* __restrict__ h,
                              float* __restrict__ agg) {
    long long idx = (long long)blockIdx.x * blockDim.x + threadIdx.x;
    const long long total = (long long)(N_EDGES + N_NODES) * (OUT_F / 4);
    if (idx >= total) return;
    int f4 = (int)(idx & (OUT_F / 4 - 1)) * 4;     // 0..60 step 4
    long long e = idx >> 4;
    int s, d;
    if (e < N_EDGES) { s = ei[e]; d = ei[N_EDGES + e]; }
    else             { s = d = (int)(e - N_EDGES); }
    float nrm = dinv[s] * dinv[d];
    v4f v = *(const v4f*)&h[(long long)s * OUT_F + f4];
    float* dst = &agg[(long long)d * OUT_F + f4];
    atomicAdd(dst + 0, v.x * nrm);
    atomicAdd(dst + 1, v.y * nrm);
    atomicAdd(dst + 2, v.z * nrm);
    atomicAdd(dst + 3, v.w * nrm);
}

// ---------------------------------------------------------------------------
extern "C" void kernel_launch(void* const* d_in, const int* in_sizes, int n_in,
                              void* d_out, int out_size, void* d_ws, size_t ws_size,
                              hipStream_t stream) {
    const float* x  = (const float*)d_in[0];
    const int*   ei = (const int*)d_in[1];   // JAX x64 disabled -> int32 [2,E]
    const float* W1 = (const float*)d_in[2];
    const float* b1 = (const float*)d_in[3];
    const float* W2 = (const float*)d_in[4];
    const float* b2 = (const float*)d_in[5];
    float* out = (float*)d_out;

    float* ws   = (float*)d_ws;
    float* dinv = ws;                                   // N
    float* h1   = dinv + N_NODES;                       // N*16
    float* agg1 = h1   + (size_t)N_NODES * HID;         // N*16 (relu'd in place)
    float* h3   = agg1 + (size_t)N_NODES * HID;         // N*64

    const int B  = 256;                                 // 8 waves / block
    const int WB = B / 32;

    long long initTot = (long long)N_NODES * OUT_F;
    gcn_init<<<(int)((initTot + B - 1) / B), B, 0, stream>>>(dinv, agg1, out);

    gcn_degree<<<(N_EDGES + B - 1) / B, B, 0, stream>>>(ei, dinv);
    gcn_rsqrt<<<(N_NODES + B - 1) / B, B, 0, stream>>>(dinv);

    int tiles1 = N_NODES / 16;                          // 6250
    gcn_gemm1_wmma<<<(tiles1 + WB - 1) / WB, B, 0, stream>>>(x, W1, h1);

    long long sc1 = (long long)(N_EDGES + N_NODES) * (HID / 4);
    gcn_scatter16<<<(int)((sc1 + B - 1) / B), B, 0, stream>>>(ei, dinv, h1, agg1);

    long long br1 = (long long)N_NODES * HID / 4;
    gcn_bias_relu<<<(int)((br1 + B - 1) / B), B, 0, stream>>>(agg1, b1, HID - 1, br1);

    int tiles2 = (N_NODES / 16) * (OUT_F / 16);         // 25000
    gcn_gemm2_wmma<<<(tiles2 + WB - 1) / WB, B, 0, stream>>>(agg1, W2, h3);

    long long sc2 = (long long)(N_EDGES + N_NODES) * (OUT_F / 4);
    gcn_scatter64<<<(int)((sc2 + B - 1) / B), B, 0, stream>>>(ei, dinv, h3, out);

    long long br2 = (long long)N_NODES * OUT_F / 4;
    gcn_bias_relu<<<(int)((br2 + B - 1) / B), B, 0, stream>>>(out, b2, OUT_F - 1, br2);
}